// DepthConv_67654324846947
// MI455X (gfx1250) — compile-verified
//
#include <hip/hip_runtime.h>
#include <stdint.h>

typedef float v4f __attribute__((ext_vector_type(4)));
typedef int   vi4 __attribute__((vector_size(16)));   // matches async builtin param type

#define AS1 __attribute__((address_space(1)))
#define AS3 __attribute__((address_space(3)))

// Fixed problem size (from the reference harness)
constexpr int N_ = 8, C_ = 128, H_ = 128, W_ = 128;
constexpr int TILE_ROWS  = 16;
constexpr int HALO_ROWS  = 18;
constexpr int ROW_STRIDE = 136;  // floats: 544 B per LDS row, keeps data 16B-aligned
constexpr int DATA_COL   = 4;    // image col c at tile col c+4; col -1 at 3; col 128 at 132
constexpr int STRIPS     = H_ / TILE_ROWS;   // 8
constexpr size_t PLANE_HW = (size_t)H_ * W_; // 16384

__global__ __launch_bounds__(256) void depthconv_dynw_kernel(
    const float* __restrict__ x,
    const float* __restrict__ w,
    float* __restrict__ out)
{
    __shared__ float tile[HALO_ROWS * ROW_STRIDE];

    const int plane = blockIdx.x >> 3;          // n*C + c   (STRIPS == 8)
    const int strip = blockIdx.x & (STRIPS - 1);
    const int row0  = strip * TILE_ROWS;        // first output row of strip

    const float* xplane = x + (size_t)plane * PLANE_HW;

    const int tid  = threadIdx.x;
    const int wave = tid >> 5;
    const int lane = tid & 31;

    // ---- stage x rows [row0-1, row0+16] into LDS ----
    // one wave32 x b128 async load == one full 512B row
    for (int r = wave; r < HALO_ROWS; r += 8) {
        const int gr = row0 - 1 + r;
        float* ldsrow = &tile[r * ROW_STRIDE + DATA_COL];
        if (gr >= 0 && gr < H_) {
            const float* grow = xplane + (size_t)gr * W_;
#if __has_builtin(__builtin_amdgcn_global_load_async_to_lds_b128)
            __builtin_amdgcn_global_load_async_to_lds_b128(
                (AS1 vi4*)((float*)grow + lane * 4),
                (AS3 vi4*)(ldsrow + lane * 4),
                /*offset=*/0, /*cpol=*/0);
#else
            v4f v = *(const v4f*)(grow + lane * 4);
            *(v4f*)(ldsrow + lane * 4) = v;
#endif
        } else {
            v4f z = {0.f, 0.f, 0.f, 0.f};
            *(v4f*)(ldsrow + lane * 4) = z;   // zero halo row outside image
        }
    }
    // zero halo columns: image col -1 (tile col 3) and image col 128 (tile col 132)
    if (tid < 2 * HALO_ROWS) {
        const int r   = tid >> 1;
        const int col = (tid & 1) ? (DATA_COL + 128) : (DATA_COL - 1);
        tile[r * ROW_STRIDE + col] = 0.f;
    }

#if __has_builtin(__builtin_amdgcn_s_wait_asynccnt)
    __builtin_amdgcn_s_wait_asynccnt(0);
#else
    asm volatile("s_wait_asynccnt 0x0" ::: "memory");
#endif
    __syncthreads();

    // ---- compute: each thread -> 8 contiguous pixels of one row ----
    const int rs = tid >> 4;           // 0..15: output row within strip
    const int c0 = (tid & 15) << 3;    // 0,8,...,120: first output column

    // 3x10 neighborhood from LDS (rows rs..rs+2, image cols c0-1 .. c0+8)
    float a[3][10];
#pragma unroll
    for (int i = 0; i < 3; ++i) {
        const float* trow = &tile[(rs + i) * ROW_STRIDE + (DATA_COL - 1) + c0];
#pragma unroll
        for (int j = 0; j < 10; ++j) a[i][j] = trow[j];
    }

    float acc[8];
#pragma unroll
    for (int j = 0; j < 8; ++j) acc[j] = 0.f;

    // weights: w[(plane*9 + k)*H*W + y*W + x], streamed once -> non-temporal
    const float* wbase = w + (size_t)plane * 9 * PLANE_HW + (size_t)(row0 + rs) * W_ + c0;
#pragma unroll
    for (int k = 0; k < 9; ++k) {
        const int ki = k / 3, kj = k % 3;
        const float* wp = wbase + (size_t)k * PLANE_HW;
        v4f w0 = __builtin_nontemporal_load((const v4f*)wp);
        v4f w1 = __builtin_nontemporal_load((const v4f*)(wp + 4));
#pragma unroll
        for (int j = 0; j < 4; ++j)
            acc[j]     = fmaf(a[ki][j + kj],     w0[j], acc[j]);
#pragma unroll
        for (int j = 0; j < 4; ++j)
            acc[4 + j] = fmaf(a[ki][4 + j + kj], w1[j], acc[4 + j]);
    }

    float* orow = out + (size_t)plane * PLANE_HW + (size_t)(row0 + rs) * W_ + c0;
    v4f o0 = {acc[0], acc[1], acc[2], acc[3]};
    v4f o1 = {acc[4], acc[5], acc[6], acc[7]};
    __builtin_nontemporal_store(o0, (v4f*)orow);
    __builtin_nontemporal_store(o1, (v4f*)(orow + 4));
}

extern "C" void kernel_launch(void* const* d_in, const int* in_sizes, int n_in,
                              void* d_out, int out_size, void* d_ws, size_t ws_size,
                              hipStream_t stream) {
    (void)in_sizes; (void)n_in; (void)out_size; (void)d_ws; (void)ws_size;
    const float* x  = (const float*)d_in[0];
    const float* wt = (const float*)d_in[1];
    float* out = (float*)d_out;

    const int blocks = N_ * C_ * STRIPS;   // 8192
    depthconv_dynw_kernel<<<dim3(blocks), dim3(256), 0, stream>>>(x, wt, out);
}